// QuantizedKVCache_52381421142177
// MI455X (gfx1250) — compile-verified
//
#include <hip/hip_runtime.h>
#include <stdint.h>

// ---------------------------------------------------------------------------
// QuantizedKVCache update — gfx1250 (MI455X)
//
// The reference quantizes new tokens, scatters into the int32 cache,
// dequantizes the WHOLE cache, then overwrites the updated slice with exact
// fp32 values — so the new tokens' quantization never reaches the output:
//    s in [pos, pos+128)  : out = val[s-pos, h, :]
//    otherwise            : out = (float(cache) - float(zp)) * scale
//
// Pure HBM stream (~512 MB, ~22 us floor @ 23.3 TB/s). To saturate 23.3 TB/s
// we need ~BW*latency ~= 23 MB of loads in flight: 8192 waves x 8-deep x
// 512 B async tiles = 32 MB. CDNA5 paths: GLOBAL_LOAD_ASYNC_TO_LDS_B128
// (ASYNCcnt, VGPR-free depth), s_wait_asynccnt, ds_load_b128, NT b128 stores.
// Scale/zp are prefetched PIPE tokens ahead into a register ring so the
// post-wait dequant has no exposed vmem latency.
// ---------------------------------------------------------------------------

#define KV_SMAX 8192
#define KV_H    32
#define KV_D    128
#define KV_SNEW 128
#define KV_NTOK (KV_SMAX * KV_H)       // 262144 tokens per tensor

#define WPB   8                        // waves per block (wave32)
#define TPB   (WPB * 32)               // 256 threads
#define NBLK  1024
#define NWAVE (NBLK * WPB)             // 8192 waves
#define TPW   (KV_NTOK / NWAVE)        // 32 tokens per wave
#define PIPE  8                        // async pipeline depth (LDS ring slots)

static_assert(KV_NTOK % NWAVE == 0, "token partition must be exact");
static_assert(TPW >= PIPE, "pipeline deeper than work per wave");
static_assert((TPW - PIPE) % PIPE == 0, "steady loop blocks exactly");
static_assert(KV_H == 32, "t>>5 / t&31 mapping assumes H==32");

typedef float v4f __attribute__((ext_vector_type(4)));
typedef int   v4i __attribute__((ext_vector_type(4)));

// ---- CDNA5 async global->LDS copy (ASYNCcnt) ------------------------------
#if defined(__HIP_DEVICE_COMPILE__)
#if __has_builtin(__builtin_amdgcn_s_wait_asynccnt)
#define WAIT_ASYNC(n) __builtin_amdgcn_s_wait_asynccnt(n)
#else
#define WAIT_ASYNC(n) asm volatile("s_wait_asynccnt " #n ::: "memory")
#endif
#else
#define WAIT_ASYNC(n) ((void)0)
#endif

__device__ __forceinline__ void async_copy_b128(const int* gsrc, int* ldst) {
#if defined(__HIP_DEVICE_COMPILE__)
#if __has_builtin(__builtin_amdgcn_global_load_async_to_lds_b128)
    // Builtin is strongly typed: (v4i AS1*, v4i AS3*, Iint offset, Iint cpol)
    __builtin_amdgcn_global_load_async_to_lds_b128(
        (__attribute__((address_space(1))) v4i*)gsrc,
        (__attribute__((address_space(3))) v4i*)ldst,
        /*offset=*/0, /*cpol=*/0);
#else
    uint32_t lds_off =
        (uint32_t)(uintptr_t)(__attribute__((address_space(3))) char*)ldst;
    asm volatile("global_load_async_to_lds_b128 %0, %1, off"
                 :: "v"(lds_off), "v"(gsrc) : "memory");
#endif
#endif
}

// ---- per-token dequant / passthrough (scale/zp pre-loaded) ----------------
__device__ __forceinline__ void process_token(
    int t, int lane, const int* wls, int slot, float sc, float zp,
    const float* __restrict__ val, int start, float* __restrict__ out)
{
    const int s = t >> 5;        // token -> seq position   (H == 32)
    const int h = t & 31;        // token -> head
    v4f r;
    if (s >= start && s < start + KV_SNEW) {
        // fresh token: output is the exact fp32 value (quantization is dead)
        const v4f* vp =
            (const v4f*)(val + ((size_t)(s - start) * KV_H + h) * (size_t)KV_D);
        r = vp[lane];
    } else {
        const v4i q = ((const v4i*)(wls + slot * KV_D))[lane];   // ds_load_b128
        r.x = ((float)q.x - zp) * sc;
        r.y = ((float)q.y - zp) * sc;
        r.z = ((float)q.z - zp) * sc;
        r.w = ((float)q.w - zp) * sc;
    }
    // touch-once output: non-temporal b128 store
    __builtin_nontemporal_store(r, (v4f*)(out + (size_t)t * KV_D) + lane);
}

// ---- main kernel: one wave == 32 tokens, 8-deep async LDS ring ------------
__global__ __launch_bounds__(TPB)
void kv_dequant_kernel(const int* __restrict__ cache,
                       const float* __restrict__ scales,
                       const int* __restrict__ zps,
                       const float* __restrict__ val,
                       const int* __restrict__ posp,
                       float* __restrict__ out)
{
    __shared__ int stage[WPB * PIPE * KV_D];   // 32 KB/block, 4 KB ring/wave

    const int lane = threadIdx.x & 31;
    const int wib  = threadIdx.x >> 5;
    const int gw   = blockIdx.x * WPB + wib;
    const int t0   = gw * TPW;
    const int start = *posp;

    int* wls = stage + wib * (PIPE * KV_D);

    float sc_ring[PIPE];   // register rings: scale/zp prefetched PIPE ahead
    float zp_ring[PIPE];

    // Prologue: fill the pipe (each async op moves one full 512 B token tile:
    // 32 lanes x 16 B) and prefetch the matching scale/zp pairs.
#pragma unroll
    for (int j = 0; j < PIPE; ++j) {
        const int t = t0 + j;
        async_copy_b128(cache + (size_t)t * KV_D + lane * 4,
                        wls + j * KV_D + lane * 4);
        sc_ring[j] = scales[t];
        zp_ring[j] = (float)zps[t];
    }

    // Steady state, blocked by PIPE so ring indices are compile-time:
    // wait for oldest tile -> consume -> refill slot + prefetch next sc/zp.
    for (int ib = 0; ib < TPW - PIPE; ib += PIPE) {
#pragma unroll
        for (int j = 0; j < PIPE; ++j) {
            WAIT_ASYNC(7);                      // oldest of 8 in flight done
            process_token(t0 + ib + j, lane, wls, j,
                          sc_ring[j], zp_ring[j], val, start, out);
            const int tn = t0 + ib + j + PIPE;  // refill the slot just freed
            async_copy_b128(cache + (size_t)tn * KV_D + lane * 4,
                            wls + j * KV_D + lane * 4);
            sc_ring[j] = scales[tn];
            zp_ring[j] = (float)zps[tn];
        }
    }

    // Drain.
    WAIT_ASYNC(0);
#pragma unroll
    for (int j = 0; j < PIPE; ++j) {
        process_token(t0 + (TPW - PIPE) + j, lane, wls, j,
                      sc_ring[j], zp_ring[j], val, start, out);
    }
}

// ---------------------------------------------------------------------------
extern "C" void kernel_launch(void* const* d_in, const int* in_sizes, int n_in,
                              void* d_out, int out_size, void* d_ws, size_t ws_size,
                              hipStream_t stream) {
    (void)in_sizes; (void)n_in; (void)out_size; (void)d_ws; (void)ws_size;

    const float* k_val    = (const float*)d_in[0];
    const float* v_val    = (const float*)d_in[1];
    const int*   k_cache  = (const int*)d_in[2];
    const int*   v_cache  = (const int*)d_in[3];
    const float* k_scales = (const float*)d_in[4];
    const float* v_scales = (const float*)d_in[5];
    const int*   k_zp     = (const int*)d_in[6];
    const int*   v_zp     = (const int*)d_in[7];
    const int*   pos      = (const int*)d_in[8];

    float* k_out = (float*)d_out;
    float* v_out = k_out + (size_t)KV_SMAX * KV_H * KV_D;

    dim3 grid(NBLK), block(TPB);
    kv_dequant_kernel<<<grid, block, 0, stream>>>(k_cache, k_scales, k_zp,
                                                  k_val, pos, k_out);
    kv_dequant_kernel<<<grid, block, 0, stream>>>(v_cache, v_scales, v_zp,
                                                  v_val, pos, v_out);
}